// KalmanFilter_29463475650752
// MI455X (gfx1250) — compile-verified
//
#include <hip/hip_runtime.h>

typedef __attribute__((ext_vector_type(2))) float v2f;
typedef __attribute__((ext_vector_type(8))) float v8f;

constexpr int kB = 256, kT = 128, kZ = 32, kY = 16, kK = 64;
constexpr int kNB = 16;               // batch tile per workgroup
constexpr int kWaves = 16;
constexpr int kThreads = kWaves * 32;

// output section offsets (floats), tuple order: mu_pred, Sigma_pred, mu_f, Sigma_f, alpha, A_t, B_t, C_t
constexpr size_t TBn  = (size_t)kT * kB;
constexpr size_t OFF_MUP = 0;
constexpr size_t OFF_SGP = OFF_MUP + TBn * kZ;
constexpr size_t OFF_MUF = OFF_SGP + TBn * kZ * kZ;
constexpr size_t OFF_SGF = OFF_MUF + TBn * kZ;
constexpr size_t OFF_AL  = OFF_SGF + TBn * kZ * kZ;
constexpr size_t OFF_A   = OFF_AL  + TBn * kK;
constexpr size_t OFF_B   = OFF_A   + TBn * kZ * kZ;
constexpr size_t OFF_C   = OFF_B   + TBn * kZ;

// LDS layout: per batch element 64+64+512+1024+32+1024+1024+1024+32+32+16 = 4848 floats
constexpr int LDS_FLOATS = kNB * 4848;   // 310,272 bytes < 320KB WGP LDS

__device__ __forceinline__ float wave_max(float v) {
  for (int o = 16; o > 0; o >>= 1) v = fmaxf(v, __shfl_xor(v, o, 32));
  return v;
}
__device__ __forceinline__ float wave_sum(float v) {
  for (int o = 16; o > 0; o >>= 1) v += __shfl_xor(v, o, 32);
  return v;
}

// One 16x16 f32 output tile of D = A(16xK) @ B(Kx16) using V_WMMA_F32_16X16X4_F32.
// A-frag: lane L holds rows M=L%16; vgpr r -> K = kk + r + 2*(L/16). B-frag mirrored.
__device__ __forceinline__ v8f gemm_tile(const float* Am, int lda, int transA,
                                         const float* Bm_, int ldb, int transB,
                                         int m0, int n0, int Kdim, int lane, v8f acc) {
  const int lh2 = (lane >> 4) << 1;   // 0 or 2
  const int ll  = lane & 15;
  for (int kk = 0; kk < Kdim; kk += 4) {
    const int ka = kk + lh2;
    v2f a, b;
    if (transA) { a.x = Am[(size_t)(ka + 0) * lda + (m0 + ll)];
                  a.y = Am[(size_t)(ka + 1) * lda + (m0 + ll)]; }
    else        { a.x = Am[(size_t)(m0 + ll) * lda + (ka + 0)];
                  a.y = Am[(size_t)(m0 + ll) * lda + (ka + 1)]; }
    if (transB) { b.x = Bm_[(size_t)(n0 + ll) * ldb + (ka + 0)];
                  b.y = Bm_[(size_t)(n0 + ll) * ldb + (ka + 1)]; }
    else        { b.x = Bm_[(size_t)(ka + 0) * ldb + (n0 + ll)];
                  b.y = Bm_[(size_t)(ka + 1) * ldb + (n0 + ll)]; }
    acc = __builtin_amdgcn_wmma_f32_16x16x4_f32(false, a, false, b, (short)0, acc, false, false);
  }
  return acc;
}

// C/D layout: vgpr v -> row M = m0 + v + 8*(lane/16), col N = n0 + lane%16
__device__ __forceinline__ void store_tile(float* D, int ldd, int m0, int n0, int lane, v8f acc) {
  const int lh = lane >> 4, ll = lane & 15;
  for (int v = 0; v < 8; ++v) D[(size_t)(m0 + v + 8 * lh) * ldd + (n0 + ll)] = acc[v];
}

__global__ __launch_bounds__(kThreads, 1)
void kvae_scan_kernel(const float* __restrict__ yg, const float* __restrict__ distg,
                      const float* __restrict__ Ag, const float* __restrict__ Bmg,
                      const float* __restrict__ Cg, float* __restrict__ out) {
  extern __shared__ float lds[];
  float* s_alpha  = lds;                       // [NB][64]  alpha at step entry
  float* s_alphaN = s_alpha  + kNB * 64;       // [NB][64]  updated alpha
  float* s_Ct     = s_alphaN + kNB * 64;       // [NB][16*32] C_t (later aliased as K_g [32][16])
  float* s_At     = s_Ct     + kNB * 512;      // [NB][32*32] A_t
  float* s_Bt     = s_At     + kNB * 1024;     // [NB][32]
  float* s_Sig    = s_Bt     + kNB * 32;       // [NB][32*32] Sigma_pred (carry)
  float* s_Sigf   = s_Sig    + kNB * 1024;     // [NB][32*32] Sigma_f
  float* s_tmp    = s_Sigf   + kNB * 1024;     // [NB][1024]: [0:512)=C*Sigma, [512:1024)=S augmented; later T2
  float* s_mu     = s_tmp    + kNB * 1024;     // [NB][32] mu_pred (carry)
  float* s_muf    = s_mu     + kNB * 32;       // [NB][32] mu_f
  float* s_inno   = s_muf    + kNB * 32;       // [NB][16]

  const int tid  = threadIdx.x;
  const int wave = tid >> 5;
  const int lane = tid & 31;
  const int b0   = blockIdx.x * kNB;
  const int b    = wave;           // per-wave batch element (kNB == kWaves)
  const int bg   = b0 + b;

  // ---------- init carry: mu0 = 0, Sigma0 = 20*I, alpha0 = softmax(dist[:,0,:]) ----------
  for (int i = tid; i < kNB * 1024; i += kThreads) {
    int r = (i >> 5) & 31, c = i & 31;
    s_Sig[i] = (r == c) ? 20.0f : 0.0f;
  }
  for (int i = tid; i < kNB * 32; i += kThreads) s_mu[i] = 0.0f;
  {
    const float* dp = distg + (size_t)bg * kT * kK;   // t = 0
    float d0_ = dp[lane], d1_ = dp[lane + 32];
    float m  = wave_max(fmaxf(d0_, d1_));
    float e0 = __expf(d0_ - m), e1 = __expf(d1_ - m);
    float es = wave_sum(e0 + e1);
    s_alpha[b * 64 + lane]      = e0 / es;
    s_alpha[b * 64 + lane + 32] = e1 / es;
  }
  __syncthreads();

  for (int t = 0; t < kT; ++t) {
    // ---- emit carry (mu_pred, Sigma_pred) ----
    for (int i = tid; i < kNB * 32; i += kThreads)
      out[OFF_MUP + ((size_t)t * kB + b0 + (i >> 5)) * 32 + (i & 31)] = s_mu[i];
    for (int i = tid; i < kNB * 1024; i += kThreads)
      out[OFF_SGP + ((size_t)t * kB + b0 + (i >> 10)) * 1024 + (i & 1023)] = s_Sig[i];

    // ---- phase 1: C_t = alpha @ C  ([16,64]x[64,512] WMMA GEMM) ----
    for (int nt = wave; nt < 32; nt += kWaves) {
      v8f acc = {};
      acc = gemm_tile(s_alpha, 64, 0, Cg, 512, 0, 0, nt * 16, 64, lane, acc);
      store_tile(s_Ct, 512, 0, nt * 16, lane, acc);
    }
    __syncthreads();
    for (int i = tid; i < kNB * 512; i += kThreads)
      out[OFF_C + ((size_t)t * kB + b0 + (i >> 9)) * 512 + (i & 511)] = s_Ct[i];
    __syncthreads();   // C_t emitted before its LDS region is reused for K_g

    // ---- phase 2: per-element Kalman update (wave b) ----
    {
      float* Ct   = s_Ct  + b * 512;     // [16][32]
      float* Sg   = s_Sig + b * 1024;    // [32][32]
      float* CtS  = s_tmp + b * 1024;    // [16][32] = C_t @ Sigma_pred
      float* Saug = CtS + 512;           // [16][32] augmented [S | I]
      float* Sinv = Saug + 16;           // right half, ld 32
      float* Kg   = Ct;                  // alias (C_t dead after S)

      for (int nt = 0; nt < 2; ++nt) {                 // CtS = Ct @ Sigma
        v8f acc = {};
        acc = gemm_tile(Ct, 32, 0, Sg, 32, 0, 0, nt * 16, 32, lane, acc);
        store_tile(CtS, 32, 0, nt * 16, lane, acc);
      }
      {                                                 // S = CtS @ Ct^T + R, build [S|I]
        v8f acc = {};
        acc = gemm_tile(CtS, 32, 0, Ct, 32, 1, 0, 0, 32, lane, acc);
        const int lh = lane >> 4, ll = lane & 15;
        for (int v = 0; v < 8; ++v) {
          int r = v + 8 * lh;
          Saug[r * 32 + ll]      = acc[v] + ((r == ll) ? 0.03f : 0.0f);
          Saug[r * 32 + 16 + ll] = (r == ll) ? 1.0f : 0.0f;
        }
      }
      // Gauss-Jordan inversion of SPD 16x16 (lane owns one of 32 augmented columns)
      for (int p = 0; p < 16; ++p) {
        float piv  = Saug[p * 32 + p];
        float rinv = 1.0f / piv;
        float rowp = Saug[p * 32 + lane] * rinv;
        Saug[p * 32 + lane] = rowp;
        for (int r = 0; r < 16; ++r) {
          if (r == p) continue;
          float f = Saug[r * 32 + p];
          Saug[r * 32 + lane] -= f * rowp;
        }
      }
      // innovation = y_t - Ct @ mu_pred
      if (lane < 16) {
        float s = 0.0f;
        for (int z = 0; z < 32; ++z) s += Ct[lane * 32 + z] * s_mu[b * 32 + z];
        s_inno[b * 16 + lane] = yg[((size_t)bg * kT + t) * 16 + lane] - s;
      }
      // K_g = Sigma@Ct^T@Sinv = CtS^T @ Sinv  ([32,16]x[16,16])
      for (int mt = 0; mt < 2; ++mt) {
        v8f acc = {};
        acc = gemm_tile(CtS, 32, 1, Sinv, 32, 0, mt * 16, 0, 16, lane, acc);
        store_tile(Kg, 16, mt * 16, 0, lane, acc);
      }
      // mu_f = mu_pred + K_g @ innov
      {
        float s = 0.0f;
        for (int yi = 0; yi < 16; ++yi) s += Kg[lane * 16 + yi] * s_inno[b * 16 + yi];
        s_muf[b * 32 + lane] = s_mu[b * 32 + lane] + s;
      }
      // Sigma_f = Sigma_pred - K_g @ (Ct@Sigma)
      for (int mt = 0; mt < 2; ++mt)
        for (int nt = 0; nt < 2; ++nt) {
          v8f acc = {};
          acc = gemm_tile(Kg, 16, 0, CtS, 32, 0, mt * 16, nt * 16, 16, lane, acc);
          const int lh = lane >> 4, ll = lane & 15;
          for (int v = 0; v < 8; ++v) {
            int r = mt * 16 + v + 8 * lh, c = nt * 16 + ll;
            s_Sigf[b * 1024 + r * 32 + c] = Sg[r * 32 + c] - acc[v];
          }
        }
      // per-wave outputs
      out[OFF_MUF + ((size_t)t * kB + bg) * 32 + lane] = s_muf[b * 32 + lane];
      for (int i = lane; i < 1024; i += 32)
        out[OFF_SGF + ((size_t)t * kB + bg) * 1024 + i] = s_Sigf[b * 1024 + i];
      // alpha update: p = softmax(dist_t)*alpha; alpha_new = p/sum(p)
      {
        const float* dp = distg + ((size_t)bg * kT + t) * kK;
        float d0_ = dp[lane], d1_ = dp[lane + 32];
        float m  = wave_max(fmaxf(d0_, d1_));
        float e0 = __expf(d0_ - m), e1 = __expf(d1_ - m);
        float es = wave_sum(e0 + e1);
        float p0 = (e0 / es) * s_alpha[b * 64 + lane];
        float p1 = (e1 / es) * s_alpha[b * 64 + lane + 32];
        float ps = wave_sum(p0 + p1);
        float a0 = p0 / ps, a1 = p1 / ps;
        s_alphaN[b * 64 + lane]      = a0;
        s_alphaN[b * 64 + lane + 32] = a1;
        out[OFF_AL + ((size_t)t * kB + bg) * 64 + lane]      = a0;
        out[OFF_AL + ((size_t)t * kB + bg) * 64 + lane + 32] = a1;
      }
    }
    __syncthreads();

    // ---- phase 3: A_t = alpha_new @ A ([16,64]x[64,1024] WMMA), B_t = alpha_new @ Bm ----
    for (int nt = wave; nt < 64; nt += kWaves) {
      v8f acc = {};
      acc = gemm_tile(s_alphaN, 64, 0, Ag, 1024, 0, 0, nt * 16, 64, lane, acc);
      store_tile(s_At, 1024, 0, nt * 16, lane, acc);
    }
    for (int i = tid; i < kNB * 32; i += kThreads) {
      int bb = i >> 5, z = i & 31;
      float s = 0.0f;
      for (int k = 0; k < 64; ++k) s += s_alphaN[bb * 64 + k] * Bmg[k * 32 + z];
      s_Bt[i] = s;
    }
    for (int i = tid; i < kNB * 64; i += kThreads) s_alpha[i] = s_alphaN[i];
    __syncthreads();
    for (int i = tid; i < kNB * 1024; i += kThreads)
      out[OFF_A + ((size_t)t * kB + b0 + (i >> 10)) * 1024 + (i & 1023)] = s_At[i];
    for (int i = tid; i < kNB * 32; i += kThreads)
      out[OFF_B + ((size_t)t * kB + b0 + (i >> 5)) * 32 + (i & 31)] = s_Bt[i];

    // ---- phase 4: predict (per-wave): Sigma_pred' = A_t Sigma_f A_t^T + Q ----
    {
      float* At = s_At   + b * 1024;
      float* T2 = s_tmp  + b * 1024;   // update-phase scratch dead
      float* Sf = s_Sigf + b * 1024;
      float* Sg = s_Sig  + b * 1024;
      for (int mt = 0; mt < 2; ++mt)
        for (int nt = 0; nt < 2; ++nt) {
          v8f acc = {};
          acc = gemm_tile(At, 32, 0, Sf, 32, 0, mt * 16, nt * 16, 32, lane, acc);
          store_tile(T2, 32, mt * 16, nt * 16, lane, acc);
        }
      for (int mt = 0; mt < 2; ++mt)
        for (int nt = 0; nt < 2; ++nt) {
          v8f acc = {};
          acc = gemm_tile(T2, 32, 0, At, 32, 1, mt * 16, nt * 16, 32, lane, acc);
          const int lh = lane >> 4, ll = lane & 15;
          for (int v = 0; v < 8; ++v) {
            int r = mt * 16 + v + 8 * lh, c = nt * 16 + ll;
            Sg[r * 32 + c] = acc[v] + ((r == c) ? 0.08f : 0.0f);
          }
        }
      // mu_pred' = A_t @ mu_f + B_t
      float s = 0.0f;
      for (int j = 0; j < 32; ++j) s += At[lane * 32 + j] * s_muf[b * 32 + j];
      s_mu[b * 32 + lane] = s + s_Bt[b * 32 + lane];
    }
    __syncthreads();
  }
}

extern "C" void kernel_launch(void* const* d_in, const int* in_sizes, int n_in,
                              void* d_out, int out_size, void* d_ws, size_t ws_size,
                              hipStream_t stream) {
  (void)in_sizes; (void)n_in; (void)d_ws; (void)ws_size; (void)out_size;
  const float* y    = (const float*)d_in[0];
  const float* dist = (const float*)d_in[1];
  const float* A    = (const float*)d_in[2];
  const float* Bm   = (const float*)d_in[3];
  const float* C    = (const float*)d_in[4];
  float* out = (float*)d_out;

  const size_t shmem = (size_t)LDS_FLOATS * sizeof(float);
  hipFuncSetAttribute(reinterpret_cast<const void*>(kvae_scan_kernel),
                      hipFuncAttributeMaxDynamicSharedMemorySize, (int)shmem);
  kvae_scan_kernel<<<dim3(kB / kNB), dim3(kThreads), shmem, stream>>>(y, dist, A, Bm, C, out);
}